// GCN_12687333392400
// MI455X (gfx1250) — compile-verified
//
#include <hip/hip_runtime.h>

typedef __attribute__((ext_vector_type(2))) float v2f;
typedef __attribute__((ext_vector_type(8))) float v8f;

#define NN 50000
#define NE 800000
#define FIN 256
#define HD 128
#define NC1 4
#define NC2 3

// ---------------- degree / normalization ----------------
__global__ void k_deg_init(float* __restrict__ deg) {
    int i = blockIdx.x * blockDim.x + threadIdx.x;
    if (i < NN) deg[i] = 1.0f;               // self-loop contribution
}

__global__ void k_deg_count(const int* __restrict__ dst, float* __restrict__ deg) {
    int e = blockIdx.x * blockDim.x + threadIdx.x;
    if (e < NE) atomicAdd(&deg[dst[e]], 1.0f);
}

__global__ void k_dinv(float* __restrict__ deg) {
    int i = blockIdx.x * blockDim.x + threadIdx.x;
    if (i < NN) deg[i] = rsqrtf(deg[i]);     // deg >= 1 always (self-loops)
}

// ---------------- fp32 WMMA GEMM: OUT[M,N] = X[M,K] @ W[K,N] ----------------
// One wave (32 lanes) computes one 16x16 tile via V_WMMA_F32_16X16X4_F32.
// A layout: lanes 0-15 -> M=lane, {v0,v1} = K=k0,k0+1 ; lanes 16-31 -> K=k0+2,k0+3
// B layout (mirrors A over K): lanes 0-15 -> N=lane, {v0,v1} = K rows k0,k0+1 ; hi half k0+2,k0+3
// C/D: VGPR j -> M = j (lanes 0-15) / j+8 (lanes 16-31), N = lane%16
__global__ void k_gemm_wmma_f32(const float* __restrict__ X, const float* __restrict__ W,
                                float* __restrict__ OUT, int M, int K, int N) {
    const int wave  = (blockIdx.x * blockDim.x + threadIdx.x) >> 5;
    const int lane  = threadIdx.x & 31;
    const int half  = lane >> 4;          // 0: lanes 0-15, 1: lanes 16-31
    const int lh    = lane & 15;
    const int tilesN = N >> 4;
    const int tm = wave / tilesN;
    const int tn = wave - tm * tilesN;
    if (tm * 16 >= M) return;             // wave-uniform guard (grid sized exactly anyway)
    const int m0 = tm << 4, n0 = tn << 4;

    v8f c = {};
    #pragma unroll 4
    for (int k0 = 0; k0 < K; k0 += 4) {
        const int ka = k0 + (half << 1);
        // A fragment: row (m0+lh), cols ka, ka+1  (contiguous -> 8B load)
        v2f a = *(const v2f*)(X + (size_t)(m0 + lh) * K + ka);
        // B fragment: rows ka, ka+1, col (n0+lh)
        const float* wp = W + (size_t)ka * N + (n0 + lh);
        v2f b;
        b.x = wp[0];
        b.y = wp[N];
        c = __builtin_amdgcn_wmma_f32_16x16x4_f32(
                /*neg_a=*/false, a, /*neg_b=*/false, b,
                /*c_mod=*/(short)0, c, /*reuse_a=*/false, /*reuse_b=*/false);
    }

    #pragma unroll
    for (int j = 0; j < 8; ++j) {
        const int row = m0 + j + (half << 3);
        OUT[(size_t)row * N + n0 + lh] = c[j];
    }
}

// ---------------- self-loop init: OUT = dinv[i]^2 * H ----------------
__global__ void k_selfloop(const float* __restrict__ Hf, const float* __restrict__ dinv,
                           float* __restrict__ OUT) {
    size_t i = (size_t)blockIdx.x * blockDim.x + threadIdx.x;
    if (i >= (size_t)NN * HD) return;
    const int node = (int)(i >> 7);
    const float dv = dinv[node];
    OUT[i] = Hf[i] * dv * dv;
}

// ---------------- edge scatter: OUT[d] += dinv[s]*dinv[d] * H[s] ----------------
// one wave32 per edge; lane owns a float4 slice of the 128-wide row
__global__ void k_edge_scatter(const float* __restrict__ Hf, const int* __restrict__ src,
                               const int* __restrict__ dst, const float* __restrict__ dinv,
                               float* __restrict__ OUT) {
    const int t = blockIdx.x * blockDim.x + threadIdx.x;
    const int e = t >> 5;
    const int lane = t & 31;
    if (e >= NE) return;
    const int s = src[e];
    const int d = dst[e];
    const float w = dinv[s] * dinv[d];
    const float4 hv = *(const float4*)(Hf + (size_t)s * HD + lane * 4);
    float* op = OUT + (size_t)d * HD + lane * 4;
    atomicAdd(op + 0, w * hv.x);
    atomicAdd(op + 1, w * hv.y);
    atomicAdd(op + 2, w * hv.z);
    atomicAdd(op + 3, w * hv.w);
}

// ---------------- bias + ReLU ----------------
__global__ void k_bias_relu(const float* __restrict__ IN, const float* __restrict__ bias,
                            float* __restrict__ OUT) {
    size_t i = (size_t)blockIdx.x * blockDim.x + threadIdx.x;
    if (i >= (size_t)NN * HD) return;
    const int f = (int)(i & (HD - 1));
    OUT[i] = fmaxf(IN[i] + bias[f], 0.0f);
}

// ---------------- layer-2 bias + heads + h output ----------------
// one wave32 per node: hf = AGG[node] + b2 ; out1 = hf@Wh1+bh1 ; out2 = hf@Wh2+bh2
__global__ void k_heads(const float* __restrict__ AGG, const float* __restrict__ b2,
                        const float* __restrict__ Wh1, const float* __restrict__ bh1,
                        const float* __restrict__ Wh2, const float* __restrict__ bh2,
                        float* __restrict__ out1, float* __restrict__ out2,
                        float* __restrict__ hout) {
    const int t = blockIdx.x * blockDim.x + threadIdx.x;
    const int node = t >> 5;
    const int lane = t & 31;
    if (node >= NN) return;

    const float4 av = *(const float4*)(AGG + (size_t)node * HD + lane * 4);
    const float4 bb = ((const float4*)b2)[lane];
    float hv[4] = { av.x + bb.x, av.y + bb.y, av.z + bb.z, av.w + bb.w };

    float4 hstore = { hv[0], hv[1], hv[2], hv[3] };
    *(float4*)(hout + (size_t)node * HD + lane * 4) = hstore;

    float acc[NC1 + NC2];
    #pragma unroll
    for (int c = 0; c < NC1 + NC2; ++c) acc[c] = 0.0f;

    #pragma unroll
    for (int j = 0; j < 4; ++j) {
        const int k = lane * 4 + j;
        #pragma unroll
        for (int c = 0; c < NC1; ++c) acc[c]       += hv[j] * Wh1[k * NC1 + c];
        #pragma unroll
        for (int c = 0; c < NC2; ++c) acc[NC1 + c] += hv[j] * Wh2[k * NC2 + c];
    }

    // wave32 reduction
    #pragma unroll
    for (int off = 16; off > 0; off >>= 1) {
        #pragma unroll
        for (int c = 0; c < NC1 + NC2; ++c)
            acc[c] += __shfl_down(acc[c], off, 32);
    }

    if (lane == 0) {
        #pragma unroll
        for (int c = 0; c < NC1; ++c) out1[(size_t)node * NC1 + c] = acc[c] + bh1[c];
        #pragma unroll
        for (int c = 0; c < NC2; ++c) out2[(size_t)node * NC2 + c] = acc[NC1 + c] + bh2[c];
    }
}

// ---------------- launcher ----------------
extern "C" void kernel_launch(void* const* d_in, const int* in_sizes, int n_in,
                              void* d_out, int out_size, void* d_ws, size_t ws_size,
                              hipStream_t stream) {
    const float* x   = (const float*)d_in[0];
    const int*   ei  = (const int*)d_in[1];
    const int*   src = ei;                 // edge_index[0]
    const int*   dst = ei + NE;            // edge_index[1]
    const float* W1  = (const float*)d_in[2];
    const float* b1  = (const float*)d_in[3];
    const float* W2  = (const float*)d_in[4];
    const float* b2  = (const float*)d_in[5];
    const float* Wh1 = (const float*)d_in[6];
    const float* bh1 = (const float*)d_in[7];
    const float* Wh2 = (const float*)d_in[8];
    const float* bh2 = (const float*)d_in[9];

    float* out1 = (float*)d_out;
    float* out2 = out1 + (size_t)NN * NC1;
    float* hout = out2 + (size_t)NN * NC2;

    // workspace: dinv (N) | bufA (N*128) | bufB (N*128)  -> ~51.5 MB
    float* dinv = (float*)d_ws;
    size_t ofs  = (((size_t)NN * sizeof(float)) + 255) & ~(size_t)255;
    float* bufA = (float*)((char*)d_ws + ofs);
    float* bufB = bufA + (size_t)NN * HD;

    const int NF_BLK   = (NN * HD + 255) / 256;        // 25000
    const int NODE_BLK = (NN + 255) / 256;
    const int EDGE_BLK = (NE + 255) / 256;
    const int SCAT_BLK = (NE * 32) / 256;              // 100000
    const int HEAD_BLK = (NN * 32 + 255) / 256;        // 6250
    const int GEMM_BLK = ((NN / 16) * (HD / 16) + 7) / 8;  // 3125 (8 waves/block)

    // normalization
    k_deg_init <<<NODE_BLK, 256, 0, stream>>>(dinv);
    k_deg_count<<<EDGE_BLK, 256, 0, stream>>>(dst, dinv);
    k_dinv     <<<NODE_BLK, 256, 0, stream>>>(dinv);

    // ---- layer 1 ----
    k_gemm_wmma_f32<<<GEMM_BLK, 256, 0, stream>>>(x, W1, bufA, NN, FIN, HD);   // h1 -> A
    k_selfloop     <<<NF_BLK,   256, 0, stream>>>(bufA, dinv, bufB);           // agg1 init -> B
    k_edge_scatter <<<SCAT_BLK, 256, 0, stream>>>(bufA, src, dst, dinv, bufB);
    k_bias_relu    <<<NF_BLK,   256, 0, stream>>>(bufB, b1, bufA);             // x2 -> A

    // ---- layer 2 ----
    k_gemm_wmma_f32<<<GEMM_BLK, 256, 0, stream>>>(bufA, W2, bufB, NN, HD, HD); // h2 -> B
    k_selfloop     <<<NF_BLK,   256, 0, stream>>>(bufB, dinv, bufA);           // agg2 init -> A (x2 dead)
    k_edge_scatter <<<SCAT_BLK, 256, 0, stream>>>(bufB, src, dst, dinv, bufA);

    // ---- bias2 + heads + h output ----
    k_heads<<<HEAD_BLK, 256, 0, stream>>>(bufA, b2, Wh1, bh1, Wh2, bh2, out1, out2, hout);
}